// FPN_81758997446674
// MI455X (gfx1250) — compile-verified
//
#include <hip/hip_runtime.h>
#include <cmath>

// CARAFE-FPN for MI455X (gfx1250), fp32 end-to-end via V_WMMA_F32_16X16X4_F32.
// Weight tiles are staged into LDS by the Tensor Data Mover (TDM), with LDS
// row padding (stride 65 floats) so A-fragment reads are bank-conflict free.

typedef float v2f __attribute__((ext_vector_type(2)));
typedef float v8f __attribute__((ext_vector_type(8)));
typedef unsigned int u32x4 __attribute__((ext_vector_type(4)));
typedef int i32x8 __attribute__((ext_vector_type(8)));
typedef int i32x4 __attribute__((ext_vector_type(4)));

#define BN_EPS 1e-5f

static __device__ __forceinline__ v8f wmma_f32_16x16x4(v2f a, v2f b, v8f c) {
    // 8 args: (neg_a, A, neg_b, B, c_mod, C, reuse_a, reuse_b)
    return __builtin_amdgcn_wmma_f32_16x16x4_f32(false, a, false, b, (short)0, c,
                                                 false, false);
}

static __device__ __forceinline__ v8f v8f_zero() {
    v8f z = {0.f, 0.f, 0.f, 0.f, 0.f, 0.f, 0.f, 0.f};
    return z;
}

// ---------------------------------------------------------------------------
// TDM: DMA an up-to-3D tile of f32 data from global into LDS (D# per ISA
// CDNA5 ISA). Tracked with TENSORcnt.
// Dim0 is the contiguous axis. tile_d2==0 -> 2D tile. pad: +1 DWORD per
// (2 << pad_int_code) DWORDs stored. clang-23 builtin takes 6 args:
// (g0, g1, g2, g3, extra_group, cpol).
// ---------------------------------------------------------------------------
static __device__ __forceinline__ void tdm_load_f32(
    unsigned lds_off_bytes, const float* gaddr,
    unsigned td0, unsigned long long s0,   // tensor_dim0, tensor_dim0_stride
    unsigned td1, unsigned long long s1,   // tensor_dim1, tensor_dim1_stride
    unsigned td2, unsigned long long s2,   // tensor_dim2, tensor_dim2_stride
    unsigned t0, unsigned t1, unsigned t2, // tile dims
    unsigned pad_int_code, unsigned pad_amt_code, bool pad_en) {
    const unsigned long long ga = (unsigned long long)(size_t)gaddr;
    u32x4 g0;
    g0.x = 1u;                                   // count=1, user descriptor
    g0.y = lds_off_bytes;                        // lds_addr (bytes)
    g0.z = (unsigned)(ga & 0xFFFFFFFFu);         // global_addr[31:0]
    g0.w = (unsigned)((ga >> 32) & 0x1FFFFFFu)   // global_addr[56:32]
           | (2u << 30);                         // type = 2 ("image")
    i32x8 g1;
    unsigned cfg = (2u << 16);                   // data_size = 4 bytes
    if (pad_en)
        cfg |= (1u << 20) | (pad_int_code << 22) | (pad_amt_code << 25);
    g1[0] = (int)cfg;                            // wg_mask=0, flags
    g1[1] = (int)((td0 & 0xFFFFu) << 16);        // atomic_addr=0 | dim0[15:0]
    g1[2] = (int)(((td0 >> 16) & 0xFFFFu) | ((td1 & 0xFFFFu) << 16));
    g1[3] = (int)(((td1 >> 16) & 0xFFFFu) | ((t0 & 0xFFFFu) << 16)); // tile0
    g1[4] = (int)((t1 & 0xFFFFu) | ((t2 & 0xFFFFu) << 16));          // tile1,2
    g1[5] = (int)(s0 & 0xFFFFFFFFu);             // dim0_stride[31:0]
    g1[6] = (int)(((s0 >> 32) & 0xFFFFu) | ((s1 & 0xFFFFu) << 16));
    g1[7] = (int)((s1 >> 16) & 0xFFFFFFFFu);     // dim1_stride[47:16]
    i32x4 g2;
    g2.x = (int)td2;                             // tensor_dim2
    g2.y = 0;                                    // tensor_dim3 / lds_incr
    g2.z = (int)(s2 & 0xFFFFFFFFu);              // dim2_stride[31:0]
    g2.w = (int)((s2 >> 32) & 0xFFFFu);          // dim2_stride hi | tile3=0
    i32x4 g3 = {0, 0, 0, 0};
    i32x8 g4 = {0, 0, 0, 0, 0, 0, 0, 0};         // extra group (unused)
    __builtin_amdgcn_tensor_load_to_lds(g0, g1, g2, g3, g4, 0);
}

// LDS flat addresses carry the wave-relative LDS offset in their low 32 bits
// (ISA §10.2 aperture mapping), so truncation yields the D# lds_addr.
static __device__ __forceinline__ unsigned lds_offset(const void* p) {
    return (unsigned)(size_t)p;
}

// ---------------------------------------------------------------------------
// 1x1 conv as GEMM: Y[b,m,p] = sum_c W[m,c] * X[b,c,p]  (+ bias | BN+ReLU)
// C_out fixed at 64. Block = 256 threads = 8 waves: 4 M-tiles x 2 N-subtiles.
// Weights TDM-staged into LDS as [o][c] tiles with row stride 65 floats.
// EPI: 0 = +bias (q0), 1 = BN(q0=g,q1=bt,q2=mean,q3=var) + ReLU
// ---------------------------------------------------------------------------
template <int EPI>
__global__ __launch_bounds__(256) void conv1x1_wmma(
    const float* __restrict__ X, const float* __restrict__ Wt,
    const float* __restrict__ q0, const float* __restrict__ q1,
    const float* __restrict__ q2, const float* __restrict__ q3,
    float* __restrict__ Y, int Cin, int HW) {
    __shared__ float Wlds[64 * 65];  // [o][c_local], padded rows

    const int tid = threadIdx.x;
    const int lane = tid & 31, wave = tid >> 5;
    const int lhalf = lane >> 4, lmod = lane & 15;
    const int mt = wave & 3, nt = wave >> 2;
    const int m0 = mt * 16;
    const int b = blockIdx.y;
    const int n = blockIdx.x * 32 + nt * 16 + lmod;  // pixel index
    const float* Xb = X + (size_t)b * Cin * HW;
    const unsigned wlds_off = lds_offset(Wlds);

    v8f acc = v8f_zero();

    for (int kb = 0; kb < Cin; kb += 64) {
        __syncthreads();  // previous tile fully consumed
        if (wave == 0) {
            // 2D tile: 64 rows (o, stride Cin) x 64 contiguous c elements.
            tdm_load_f32(wlds_off, Wt + kb,
                         /*td0*/ (unsigned)(Cin - kb), /*s0*/ (unsigned)Cin,
                         /*td1*/ 64u, /*s1*/ 0ull,
                         /*td2*/ 0u, /*s2*/ 0ull,
                         /*t0*/ 64u, /*t1*/ 64u, /*t2*/ 0u,
                         /*pad_int (64 DW)*/ 5u, /*pad_amt (1 DW)*/ 0u, true);
            __builtin_amdgcn_s_wait_tensorcnt(0);
        }
        __syncthreads();  // tile visible to all waves

#pragma unroll 4
        for (int k = 0; k < 64; k += 4) {
            const int kk = k + 2 * lhalf;  // per-lane K base (ISA A/B layout)
            v2f a, bf;
            a.x = Wlds[(m0 + lmod) * 65 + kk + 0];
            a.y = Wlds[(m0 + lmod) * 65 + kk + 1];
            const size_t gk = (size_t)(kb + kk) * HW + n;
            bf.x = Xb[gk];
            bf.y = Xb[gk + HW];
            __builtin_prefetch(Xb + gk + 8 * (size_t)HW, 0, 1);
            acc = wmma_f32_16x16x4(a, bf, acc);
        }
    }

#pragma unroll
    for (int r = 0; r < 8; ++r) {
        const int m = m0 + r + 8 * lhalf;  // C/D layout: M = vgpr + 8*(lane>=16)
        float v = acc[r];
        if (EPI == 0) {
            v += q0[m];
        } else {
            const float s = q0[m] * rsqrtf(q3[m] + BN_EPS);
            v = v * s + (q1[m] - q2[m] * s);
            v = v > 0.f ? v : 0.f;
        }
        Y[((size_t)b * 64 + m) * HW + n] = v;
    }
}

// ---------------------------------------------------------------------------
// 3x3 conv (pad=1, Cin=64) as implicit GEMM. K loop = 9 taps x 16 chunks of 4.
// MT = number of 16-row M tiles (4 -> Cout 64, 7 -> Cout 100 padded to 112).
// Per-tap weight slice TDM-staged as a 3D gather (dim1 = c stride 9,
// dim2 = o stride 576); tile_dim2 = MT*16 > Cout zero-fills padded rows via
// the TDM OOB rule. EPI: 0 = +bias (q0), 1 = BN only.
// ---------------------------------------------------------------------------
template <int MT, int EPI>
__global__ __launch_bounds__(256) void conv3x3_wmma(
    const float* __restrict__ X, const float* __restrict__ Wt,
    const float* __restrict__ q0, const float* __restrict__ q1,
    const float* __restrict__ q2, const float* __restrict__ q3,
    float* __restrict__ Y, int Cout, int H, int Wd) {
    constexpr int MREG = MT * 16;
    __shared__ float Wlds[MREG * 65];  // [o][c], padded rows

    const int tid = threadIdx.x;
    const int lane = tid & 31, wave = tid >> 5;
    const int lhalf = lane >> 4, lmod = lane & 15;
    const int HW = H * Wd;
    const int b = blockIdx.y;
    const int p = blockIdx.x * 128 + wave * 16 + lmod;
    const bool pvalid = p < HW;
    const int pc = pvalid ? p : 0;
    const int ph = pc / Wd, pw = pc % Wd;
    const float* Xb = X + (size_t)b * 64 * HW;
    const unsigned wlds_off = lds_offset(Wlds);

    v8f acc[MT];
#pragma unroll
    for (int i = 0; i < MT; ++i) acc[i] = v8f_zero();

    for (int t = 0; t < 9; ++t) {
        const int ty = t / 3, tx = t % 3;
        const int dy = ty - 1, dx = tx - 1;
        __syncthreads();  // previous tap tile consumed
        if (wave == 0) {
            // 3D: dim0 = 1 elem, dim1 = 64 c (stride 9), dim2 = o (stride 576).
            tdm_load_f32(wlds_off, Wt + (ty * 3 + tx),
                         /*td0*/ 1u, /*s0*/ 1ull,
                         /*td1*/ 64u, /*s1*/ 9ull,
                         /*td2*/ (unsigned)Cout, /*s2*/ 576ull,
                         /*t0*/ 1u, /*t1*/ 64u, /*t2*/ (unsigned)MREG,
                         /*pad_int (64 DW)*/ 5u, /*pad_amt (1 DW)*/ 0u, true);
            __builtin_amdgcn_s_wait_tensorcnt(0);
        }
        __syncthreads();

        const int hh = ph + dy, ww = pw + dx;
        const bool vld = pvalid && (hh >= 0) && (hh < H) && (ww >= 0) && (ww < Wd);
        const int soff = vld ? hh * Wd + ww : 0;  // clamped; data masked to 0

#pragma unroll 4
        for (int kc = 0; kc < 64; kc += 4) {
            const int kk = kc + 2 * lhalf;
            v2f bf;
            bf.x = vld ? Xb[(size_t)(kk + 0) * HW + soff] : 0.f;
            bf.y = vld ? Xb[(size_t)(kk + 1) * HW + soff] : 0.f;
#pragma unroll
            for (int m = 0; m < MT; ++m) {
                v2f a;
                a.x = Wlds[(m * 16 + lmod) * 65 + kk + 0];
                a.y = Wlds[(m * 16 + lmod) * 65 + kk + 1];
                acc[m] = wmma_f32_16x16x4(a, bf, acc[m]);
            }
        }
    }

    if (!pvalid) return;
#pragma unroll
    for (int mt = 0; mt < MT; ++mt) {
#pragma unroll
        for (int r = 0; r < 8; ++r) {
            const int m = mt * 16 + r + 8 * lhalf;
            if (m >= Cout) continue;
            float v = acc[mt][r];
            if (EPI == 0) {
                v += q0[m];
            } else {
                const float s = q0[m] * rsqrtf(q3[m] + BN_EPS);
                v = v * s + (q1[m] - q2[m] * s);
            }
            Y[((size_t)b * Cout + m) * HW + p] = v;
        }
    }
}

// ---------------------------------------------------------------------------
// Fused pixel-shuffle + softmax(25) + CARAFE 5x5 dilated apply, += into the
// finer lateral map. E: post-BN enc logits (b,100,h,w). Xc: coarse (b,64,h,w).
// Yf: fine (b,64,2h,2w), accumulated in place.
// ---------------------------------------------------------------------------
__global__ __launch_bounds__(256) void carafe_apply(
    const float* __restrict__ E, const float* __restrict__ Xc,
    float* __restrict__ Yf, int B_, int h, int w) {
    const int w2 = 2 * w, h2 = 2 * h;
    const int hw = h * w, hw2 = h2 * w2;
    const int q = blockIdx.x * blockDim.x + threadIdx.x;
    if (q >= B_ * hw2) return;
    const int b = q / hw2;
    const int rem = q - b * hw2;
    const int h_ = rem / w2, w_ = rem - h_ * w2;
    const int hs = h_ >> 1, ws = w_ >> 1;
    const int sub = (h_ & 1) * 2 + (w_ & 1);

    const float* Eb = E + (size_t)b * 100 * hw + (size_t)hs * w + ws;
    float m[25];
    float mx = -1e30f;
#pragma unroll
    for (int k = 0; k < 25; ++k) {
        m[k] = Eb[(size_t)(k * 4 + sub) * hw];
        mx = fmaxf(mx, m[k]);
    }
    float s = 0.f;
#pragma unroll
    for (int k = 0; k < 25; ++k) {
        m[k] = __expf(m[k] - mx);
        s += m[k];
    }
    const float inv = 1.f / s;

    int off[25];
#pragma unroll
    for (int k = 0; k < 25; ++k) {
        const int i = k / 5, j = k % 5;
        const int r = hs + i - 2, c = ws + j - 2;
        const bool v = (r >= 0) & (r < h) & (c >= 0) & (c < w);
        off[k] = v ? r * w + c : 0;
        m[k] = v ? m[k] * inv : 0.f;
    }

    const float* Xb = Xc + (size_t)b * 64 * hw;
    float* Yb = Yf + (size_t)b * 64 * hw2 + (size_t)h_ * w2 + w_;
    for (int c = 0; c < 64; ++c) {
        const float* src = Xb + (size_t)c * hw;
        float a = 0.f;
#pragma unroll
        for (int k = 0; k < 25; ++k) a += m[k] * src[off[k]];
        Yb[(size_t)c * hw2] += a;
    }
}

// ---------------------------------------------------------------------------
// Host-side orchestration. Input order = setup_inputs() dict order.
// ---------------------------------------------------------------------------
extern "C" void kernel_launch(void* const* d_in, const int* in_sizes, int n_in,
                              void* d_out, int out_size, void* d_ws,
                              size_t ws_size, hipStream_t stream) {
    (void)in_sizes; (void)n_in; (void)out_size; (void)ws_size;
    const float* x0 = (const float*)d_in[0];
    const float* lat0_w = (const float*)d_in[1];
    const float* lat0_b = (const float*)d_in[2];
    const float* fpn0_w = (const float*)d_in[3];
    const float* fpn0_b = (const float*)d_in[4];
    const float* x1 = (const float*)d_in[5];
    const float* lat1_w = (const float*)d_in[6];
    const float* lat1_b = (const float*)d_in[7];
    const float* fpn1_w = (const float*)d_in[8];
    const float* fpn1_b = (const float*)d_in[9];
    const float* x2 = (const float*)d_in[10];
    const float* lat2_w = (const float*)d_in[11];
    const float* lat2_b = (const float*)d_in[12];
    const float* fpn2_w = (const float*)d_in[13];
    const float* fpn2_b = (const float*)d_in[14];
    const float* u0_comp_w = (const float*)d_in[15];
    const float* u0_enc_w  = (const float*)d_in[16];
    const float* u0_comp_g = (const float*)d_in[17];
    const float* u0_comp_bt= (const float*)d_in[18];
    const float* u0_comp_m = (const float*)d_in[19];
    const float* u0_comp_v = (const float*)d_in[20];
    const float* u0_enc_g  = (const float*)d_in[21];
    const float* u0_enc_bt = (const float*)d_in[22];
    const float* u0_enc_m  = (const float*)d_in[23];
    const float* u0_enc_v  = (const float*)d_in[24];
    const float* u1_comp_w = (const float*)d_in[25];
    const float* u1_enc_w  = (const float*)d_in[26];
    const float* u1_comp_g = (const float*)d_in[27];
    const float* u1_comp_bt= (const float*)d_in[28];
    const float* u1_comp_m = (const float*)d_in[29];
    const float* u1_comp_v = (const float*)d_in[30];
    const float* u1_enc_g  = (const float*)d_in[31];
    const float* u1_enc_bt = (const float*)d_in[32];
    const float* u1_enc_m  = (const float*)d_in[33];
    const float* u1_enc_v  = (const float*)d_in[34];

    float* out = (float*)d_out;
    float* ws = (float*)d_ws;

    const int B = 4;
    // Workspace layout (floats): 7,680,000 total (~30.7 MB)
    float* lat0 = ws;                  // 4*64*96*160 = 3,932,160
    float* lat1 = lat0 + 3932160;      // 4*64*48*80  =   983,040
    float* lat2 = lat1 + 983040;       // 4*64*24*40  =   245,760
    float* comp = lat2 + 245760;       // shared, max  =   983,040
    float* enc  = comp + 983040;       // shared, max  = 1,536,000

    const dim3 blk(256);

    // Lateral 1x1 convs (+bias)
    conv1x1_wmma<0><<<dim3(15360 / 32, B), blk, 0, stream>>>(
        x0, lat0_w, lat0_b, lat0_b, lat0_b, lat0_b, lat0, 128, 15360);
    conv1x1_wmma<0><<<dim3(3840 / 32, B), blk, 0, stream>>>(
        x1, lat1_w, lat1_b, lat1_b, lat1_b, lat1_b, lat1, 256, 3840);
    conv1x1_wmma<0><<<dim3(960 / 32, B), blk, 0, stream>>>(
        x2, lat2_w, lat2_b, lat2_b, lat2_b, lat2_b, lat2, 512, 960);

    // CARAFE u1: lat2 (24x40) -> upsample, add into lat1 (48x80)
    conv1x1_wmma<1><<<dim3(960 / 32, B), blk, 0, stream>>>(
        lat2, u1_comp_w, u1_comp_g, u1_comp_bt, u1_comp_m, u1_comp_v, comp, 64, 960);
    conv3x3_wmma<7, 1><<<dim3((960 + 127) / 128, B), blk, 0, stream>>>(
        comp, u1_enc_w, u1_enc_g, u1_enc_bt, u1_enc_m, u1_enc_v, enc, 100, 24, 40);
    carafe_apply<<<dim3((B * 48 * 80 + 255) / 256), blk, 0, stream>>>(
        enc, lat2, lat1, B, 24, 40);

    // CARAFE u0: lat1 (48x80) -> upsample, add into lat0 (96x160)
    conv1x1_wmma<1><<<dim3(3840 / 32, B), blk, 0, stream>>>(
        lat1, u0_comp_w, u0_comp_g, u0_comp_bt, u0_comp_m, u0_comp_v, comp, 64, 3840);
    conv3x3_wmma<7, 1><<<dim3((3840 + 127) / 128, B), blk, 0, stream>>>(
        comp, u0_enc_w, u0_enc_g, u0_enc_bt, u0_enc_m, u0_enc_v, enc, 100, 48, 80);
    carafe_apply<<<dim3((B * 96 * 160 + 255) / 256), blk, 0, stream>>>(
        enc, lat1, lat0, B, 48, 80);

    // Output 3x3 convs (+bias) straight into d_out (concatenated tuple)
    conv3x3_wmma<4, 0><<<dim3((15360 + 127) / 128, B), blk, 0, stream>>>(
        lat0, fpn0_w, fpn0_b, fpn0_b, fpn0_b, fpn0_b, out, 64, 96, 160);
    conv3x3_wmma<4, 0><<<dim3((3840 + 127) / 128, B), blk, 0, stream>>>(
        lat1, fpn1_w, fpn1_b, fpn1_b, fpn1_b, fpn1_b, out + 3932160, 64, 48, 80);
    conv3x3_wmma<4, 0><<<dim3((960 + 127) / 128, B), blk, 0, stream>>>(
        lat2, fpn2_w, fpn2_b, fpn2_b, fpn2_b, fpn2_b, out + 3932160 + 983040, 64,
        24, 40);
}